// SpikeRefractoryIncidence_24275155157343
// MI455X (gfx1250) — compile-verified
//
#include <hip/hip_runtime.h>
#include <hip/hip_bf16.h>
#include <math.h>

// Problem constants (match reference): B=8, N=2048, D=128, V=32
#define BB 8
#define NN 2048
#define DD 128
#define VV 32
#define CELLS (BB * NN)      // 16384
#define TWO_D (2 * DD)       // 256
#define EOUT (DD / 4)        // 32

typedef __attribute__((ext_vector_type(2))) float v2f;
typedef __attribute__((ext_vector_type(8))) float v8f;

// ---------------------------------------------------------------------------
// Kernel 1: per-variable mean via LDS float atomics (ds_add_f32).
// One workgroup per batch. vmean[b][v][d] = sum(obs where idx==v)/max(count,1)
// ---------------------------------------------------------------------------
__global__ void k1_varmean(const float* __restrict__ obs,
                           const int* __restrict__ idx,
                           float* __restrict__ vmean) {
    __shared__ float vsum[VV * DD];   // 16 KB
    __shared__ float vcnt[VV];
    const int b = blockIdx.x;
    const int tid = threadIdx.x;
    const int bdim = blockDim.x;

    for (int p = tid; p < VV * DD; p += bdim) vsum[p] = 0.0f;
    for (int p = tid; p < VV; p += bdim) vcnt[p] = 0.0f;
    __syncthreads();

    const int* idxB = idx + b * NN;
    const float* obsB = obs + (size_t)b * NN * DD;

    for (int n = tid; n < NN; n += bdim)
        atomicAdd(&vcnt[idxB[n]], 1.0f);

    for (int p = tid; p < NN * DD; p += bdim) {
        const int n = p >> 7;           // / DD
        const int d = p & (DD - 1);
        atomicAdd(&vsum[idxB[n] * DD + d], obsB[p]);
    }
    __syncthreads();

    float* out = vmean + (size_t)b * VV * DD;
    for (int p = tid; p < VV * DD; p += bdim) {
        const float cnt = fmaxf(vcnt[p >> 7], 1.0f);
        out[p] = vsum[p] / cnt;
    }
}

// ---------------------------------------------------------------------------
// Kernel 2: membrane gate per cell.
// membrane = obs . W_mem[0:128] + (obs - vmean) . W_mem[128:256] + b_mem
// g_raw = sigmoid(membrane) * mask
// ---------------------------------------------------------------------------
__global__ void k2_gate(const float* __restrict__ obs,
                        const int* __restrict__ idx,
                        const float* __restrict__ vmean,
                        const float* __restrict__ mask,
                        const float* __restrict__ W_mem,
                        const float* __restrict__ b_mem,
                        float* __restrict__ g_raw) {
    const int cell = blockIdx.x * blockDim.x + threadIdx.x;
    if (cell >= CELLS) return;
    const int b = cell >> 11;           // / NN
    const int v = idx[cell];
    const float* o = obs + (size_t)cell * DD;
    const float* c = vmean + ((size_t)b * VV + v) * DD;
    float acc = 0.0f;
    for (int d = 0; d < DD; ++d) {
        const float ob = o[d];
        acc = fmaf(ob, W_mem[d], acc);
        acc = fmaf(ob - c[d], W_mem[DD + d], acc);
    }
    const float memb = acc + b_mem[0];
    const float g = (1.0f / (1.0f + expf(-memb))) * mask[cell];
    g_raw[cell] = g;
}

// ---------------------------------------------------------------------------
// Kernel 3: event projection GEMM via V_WMMA_F32_16X16X4_F32.
// e_pre[row][j] = sum_k mem_in[row][k] * W_evt[k][j] + b_evt[j]
// where mem_in[row] = [obs[row], obs[row]-vmean[b][v]]  (built on the fly).
// Each wave: 16 rows x 32 cols (two 16x16 f32 accumulators).
// K-loop is split into two branch-free halves (obs half / deviation half)
// so no divergent control flow appears between the WMMA issues.
// ---------------------------------------------------------------------------
__global__ void k3_event_gemm(const float* __restrict__ obs,
                              const int* __restrict__ idx,
                              const float* __restrict__ vmean,
                              const float* __restrict__ W_evt,
                              const float* __restrict__ b_evt,
                              float* __restrict__ e_pre) {
    const int wave = (blockIdx.x * blockDim.x + threadIdx.x) >> 5;  // 0..1023
    const int lane = threadIdx.x & 31;
    const int rowBase = wave * 16;
    const int M = lane & 15;
    const int row = rowBase + M;           // flat cell index 0..16383
    const int b = row >> 11;
    const int v = idx[row];
    const float* obsRow = obs + (size_t)row * DD;
    const float* ctxRow = vmean + ((size_t)b * VV + v) * DD;
    const int khalf = (lane >> 4) * 2;     // lanes 0-15 -> K+0..1, 16-31 -> K+2..3
    const int Ncol = lane & 15;

    v8f c0 = {};
    v8f c1 = {};

    // ---- First half: A = obs[row][k], k in [0,128) ----
#pragma unroll 4
    for (int k0 = 0; k0 < DD; k0 += 4) {
        const int kb = k0 + khalf;
        v2f a;
        a.x = obsRow[kb];
        a.y = obsRow[kb + 1];
        v2f b0, b1;
        b0.x = W_evt[kb * EOUT + Ncol];
        b0.y = W_evt[(kb + 1) * EOUT + Ncol];
        b1.x = W_evt[kb * EOUT + Ncol + 16];
        b1.y = W_evt[(kb + 1) * EOUT + Ncol + 16];
        c0 = __builtin_amdgcn_wmma_f32_16x16x4_f32(false, a, false, b0,
                                                   (short)0, c0, false, false);
        c1 = __builtin_amdgcn_wmma_f32_16x16x4_f32(false, a, false, b1,
                                                   (short)0, c1, false, false);
    }

    // ---- Second half: A = obs[row][k] - vmean[b][v][k], W rows 128..255 ----
#pragma unroll 4
    for (int k0 = 0; k0 < DD; k0 += 4) {
        const int kb = k0 + khalf;         // index into obs/vmean
        const int kw = kb + DD;            // index into W_evt rows
        v2f a;
        a.x = obsRow[kb] - ctxRow[kb];
        a.y = obsRow[kb + 1] - ctxRow[kb + 1];
        v2f b0, b1;
        b0.x = W_evt[kw * EOUT + Ncol];
        b0.y = W_evt[(kw + 1) * EOUT + Ncol];
        b1.x = W_evt[kw * EOUT + Ncol + 16];
        b1.y = W_evt[(kw + 1) * EOUT + Ncol + 16];
        c0 = __builtin_amdgcn_wmma_f32_16x16x4_f32(false, a, false, b0,
                                                   (short)0, c0, false, false);
        c1 = __builtin_amdgcn_wmma_f32_16x16x4_f32(false, a, false, b1,
                                                   (short)0, c1, false, false);
    }

    // C/D layout: VGPR r -> M = r + 8*(lane/16), N = lane%16
    const int mAdd = (lane >> 4) * 8;
    const float be0 = b_evt[Ncol];
    const float be1 = b_evt[Ncol + 16];
#pragma unroll
    for (int r = 0; r < 8; ++r) {
        const int outRow = rowBase + r + mAdd;
        e_pre[(size_t)outRow * EOUT + Ncol] = c0[r] + be0;
        e_pre[(size_t)outRow * EOUT + Ncol + 16] = c1[r] + be1;
    }
}

// ---------------------------------------------------------------------------
// Kernel 4: refractory inhibition (O(N^2) per batch, LDS-tiled) + outputs.
// inh[n] = mask[n] * sum_m [idx_m==idx_n][t_m<t_n] exp(-(t_n-t_m)/tau) g_raw[m]
// g_refr = 1 - tanh(alpha*inh); g_n = g_raw*g_refr; e_n = e_pre*g_raw*g_refr
// ---------------------------------------------------------------------------
__global__ void k4_refractory(const float* __restrict__ time_norm,
                              const float* __restrict__ mask,
                              const int* __restrict__ idx,
                              const float* __restrict__ g_raw,
                              const float* __restrict__ e_pre,
                              const float* __restrict__ log_tau,
                              const float* __restrict__ log_alpha,
                              float* __restrict__ out_g,
                              float* __restrict__ out_e) {
    __shared__ float tl[NN];   // 8 KB
    __shared__ float gl[NN];   // 8 KB
    __shared__ int vl[NN];     // 8 KB
    const int b = blockIdx.x >> 3;       // / 8 chunks
    const int chunk = blockIdx.x & 7;
    const int tid = threadIdx.x;
    const int base = b * NN;

    for (int m = tid; m < NN; m += blockDim.x) {
        tl[m] = time_norm[base + m];
        gl[m] = g_raw[base + m];
        vl[m] = idx[base + m];
    }
    __syncthreads();

    const int n = chunk * 256 + tid;
    const float tn = tl[n];
    const int vn = vl[n];
    const float tau = fmaxf(expf(log_tau[vn]), 1e-6f);
    const float inv_tau = 1.0f / tau;
    const float alpha = expf(log_alpha[vn]);

    float inh = 0.0f;
    for (int m = 0; m < NN; ++m) {
        const float dt = tn - tl[m];
        if (vl[m] == vn && dt > 0.0f)
            inh += expf(-dt * inv_tau) * gl[m];  // gl already includes mask_m
    }
    inh *= mask[base + n];                       // valid_pair's mask_n factor

    const float g_refr = 1.0f - tanhf(alpha * inh);
    const float scale = gl[n] * g_refr;          // g_raw * g_refr
    out_g[base + n] = scale;

    const float* ep = e_pre + (size_t)(base + n) * EOUT;
    float* oe = out_e + (size_t)(base + n) * EOUT;
#pragma unroll
    for (int j = 0; j < EOUT; ++j)
        oe[j] = ep[j] * scale;
}

// ---------------------------------------------------------------------------
extern "C" void kernel_launch(void* const* d_in, const int* in_sizes, int n_in,
                              void* d_out, int out_size, void* d_ws, size_t ws_size,
                              hipStream_t stream) {
    (void)in_sizes; (void)n_in; (void)out_size; (void)ws_size;

    const float* obs       = (const float*)d_in[0];   // (B,N,D)
    const float* mask      = (const float*)d_in[1];   // (B,N)
    /* d_in[2] = incidence matrix (unused; idx is the same info) */
    const int*   idx       = (const int*)d_in[3];     // (B,N) int32
    const float* time_norm = (const float*)d_in[4];   // (B,N)
    const float* W_mem     = (const float*)d_in[5];   // (256,1)
    const float* b_mem     = (const float*)d_in[6];   // (1,)
    const float* W_evt     = (const float*)d_in[7];   // (256,32)
    const float* b_evt     = (const float*)d_in[8];   // (32,)
    const float* log_tau   = (const float*)d_in[9];   // (32,)
    const float* log_alpha = (const float*)d_in[10];  // (32,)

    // Workspace layout (floats): vmean | g_raw | e_pre  (~2.3 MB total)
    float* vmean = (float*)d_ws;                      // B*V*D = 32768
    float* g_raw = vmean + (size_t)BB * VV * DD;      // B*N   = 16384
    float* e_pre = g_raw + (size_t)CELLS;             // B*N*32 = 524288

    float* out_g = (float*)d_out;                     // (B,N)
    float* out_e = out_g + CELLS;                     // (B,N,32)

    k1_varmean<<<BB, 256, 0, stream>>>(obs, idx, vmean);
    k2_gate<<<CELLS / 256, 256, 0, stream>>>(obs, idx, vmean, mask, W_mem, b_mem, g_raw);
    // 1024 waves of 16 rows each: 256 blocks x 128 threads (4 waves/block)
    k3_event_gemm<<<256, 128, 0, stream>>>(obs, idx, vmean, W_evt, b_evt, e_pre);
    k4_refractory<<<BB * 8, 256, 0, stream>>>(time_norm, mask, idx, g_raw, e_pre,
                                              log_tau, log_alpha, out_g, out_e);
}